// IsLandLoss_28561532519009
// MI455X (gfx1250) — compile-verified
//
#include <hip/hip_runtime.h>
#include <hip/hip_bf16.h>

typedef float v2f __attribute__((ext_vector_type(2)));
typedef float v8f __attribute__((ext_vector_type(8)));

#define B_SZ 16384
#define C_SZ 2048
#define D_SZ 2048
#define TLAMBDA 0.01f

// ws layout (in floats):
//  [0,     2048)   inv_norm[C]
//  [2048,  4096)   s[D]
//  [4096,  36864)  s_part[16][D]
//  [36864, 38912)  cnt[C]      (int)
//  [38912, 40960)  isl_part[C]
//  [40960, 57344)  cl_part[B]
// total 57344 floats = 224 KB

__global__ void init_cnt_k(int* __restrict__ cnt) {
    int i = blockIdx.x * 256 + threadIdx.x;
    if (i < C_SZ) cnt[i] = 0;
}

// inv_norm[c] = 1/||centers[c]||  -- one block per row
__global__ void rownorm_k(const float* __restrict__ centers,
                          float* __restrict__ inv_norm) {
    __shared__ float red[256];
    int c = blockIdx.x;
    const float* row = centers + (size_t)c * D_SZ;
    float acc = 0.f;
    for (int i = threadIdx.x; i < D_SZ; i += 256) {
        float v = row[i];
        acc += v * v;
    }
    red[threadIdx.x] = acc;
    __syncthreads();
    for (int off = 128; off > 0; off >>= 1) {
        if (threadIdx.x < off) red[threadIdx.x] += red[threadIdx.x + off];
        __syncthreads();
    }
    if (threadIdx.x == 0) inv_norm[c] = 1.0f / sqrtf(red[0]);
}

// s_part[chunk][d] = sum over 128 rows of cn[c][d]   grid (D/256, 16)
__global__ void colsum_k(const float* __restrict__ centers,
                         const float* __restrict__ inv_norm,
                         float* __restrict__ s_part) {
    int d  = blockIdx.x * 256 + threadIdx.x;
    int c0 = blockIdx.y * (C_SZ / 16);
    float acc = 0.f;
    for (int c = c0; c < c0 + C_SZ / 16; ++c)
        acc += centers[(size_t)c * D_SZ + d] * inv_norm[c];
    s_part[blockIdx.y * D_SZ + d] = acc;
}

__global__ void sreduce_k(const float* __restrict__ s_part,
                          float* __restrict__ s) {
    int d = blockIdx.x * 256 + threadIdx.x;
    float acc = 0.f;
    for (int k = 0; k < 16; ++k) acc += s_part[k * D_SZ + d];
    s[d] = acc;
}

// histogram of labels (integer atomics: deterministic)
__global__ void hist_k(const int* __restrict__ label, int* __restrict__ cnt) {
    int i = blockIdx.x * 256 + threadIdx.x;
    atomicAdd(&cnt[label[i]], 1);
}

// isl_part[c] = cnt[c] * (cn[c]·s + (C-2))  via v_wmma_f32_16x16x4_f32.
// One wave (32 threads) per 16 classes; B = s replicated across 16 columns.
// A layout (ISA 7.12.2): lanes 0-15 hold K=0,1 in VGPR0,1; lanes 16-31 K=2,3.
__global__ void classdot_k(const float* __restrict__ centers,
                           const float* __restrict__ inv_norm,
                           const float* __restrict__ s,
                           const int* __restrict__ cnt,
                           float* __restrict__ isl_part) {
    int lane = threadIdx.x;             // 0..31, one full wave
    int m0   = blockIdx.x * 16;
    int row  = m0 + (lane & 15);
    int kb   = (lane >> 4) * 2;         // 0 or 2
    float rn = inv_norm[row];
    const v2f* crow2 = (const v2f*)(centers + (size_t)row * D_SZ);
    const v2f* s2    = (const v2f*)s;

    v8f acc = {};
    for (int k = 0; k < D_SZ; k += 4) {
        int p = (k + kb) >> 1;          // 8-byte aligned v2f index
        v2f a = crow2[p] * rn;          // A fragment (normalized centers)
        v2f b = s2[p];                  // B fragment (s broadcast to all N)
        acc = __builtin_amdgcn_wmma_f32_16x16x4_f32(
            false, a, false, b, (short)0, acc, false, false);
    }
    // D layout: lanes 0-15: VGPR v -> M=v ; lanes 16-31: VGPR v -> M=v+8.
    // Every column N holds the same dot for its row; read column N=0 (lane 0)
    // and N=0 of the upper half (lane 16).
    if (lane == 0) {
        for (int v = 0; v < 8; ++v)
            isl_part[m0 + v] = (float)cnt[m0 + v] * (acc[v] + (float)(C_SZ - 2));
    } else if (lane == 16) {
        for (int v = 0; v < 8; ++v)
            isl_part[m0 + 8 + v] = (float)cnt[m0 + 8 + v] * (acc[v] + (float)(C_SZ - 2));
    }
}

// cl_part[b] = sum_d (feat[b,d] - centers[label[b],d])^2  -- one block per row
__global__ void closs_k(const float* __restrict__ feat,
                        const float* __restrict__ centers,
                        const int* __restrict__ label,
                        float* __restrict__ cl_part) {
    __shared__ float red[256];
    int b   = blockIdx.x;
    int lab = label[b];
    const float4* f = (const float4*)(feat + (size_t)b * D_SZ);
    const float4* c = (const float4*)(centers + (size_t)lab * D_SZ);
    float acc = 0.f;
    for (int i = threadIdx.x; i < D_SZ / 4; i += 256) {
        float4 fv = f[i], cv = c[i];
        float dx = fv.x - cv.x, dy = fv.y - cv.y;
        float dz = fv.z - cv.z, dw = fv.w - cv.w;
        acc += dx * dx + dy * dy + dz * dz + dw * dw;
    }
    red[threadIdx.x] = acc;
    __syncthreads();
    for (int off = 128; off > 0; off >>= 1) {
        if (threadIdx.x < off) red[threadIdx.x] += red[threadIdx.x + off];
        __syncthreads();
    }
    if (threadIdx.x == 0) cl_part[b] = red[0];
}

// out[0] = TLAMBDA * sum(isl_part) + 0.5/B * sum(cl_part)
__global__ void final_k(const float* __restrict__ cl_part,
                        const float* __restrict__ isl_part,
                        float* __restrict__ out) {
    __shared__ float red[1024];
    float cl = 0.f;
    for (int i = threadIdx.x; i < B_SZ; i += 1024) cl += cl_part[i];
    float isl = 0.f;
    for (int i = threadIdx.x; i < C_SZ; i += 1024) isl += isl_part[i];
    red[threadIdx.x] = cl * (0.5f / (float)B_SZ) + isl * TLAMBDA;
    __syncthreads();
    for (int off = 512; off > 0; off >>= 1) {
        if (threadIdx.x < off) red[threadIdx.x] += red[threadIdx.x + off];
        __syncthreads();
    }
    if (threadIdx.x == 0) out[0] = red[0];
}

extern "C" void kernel_launch(void* const* d_in, const int* in_sizes, int n_in,
                              void* d_out, int out_size, void* d_ws, size_t ws_size,
                              hipStream_t stream) {
    const int*   label   = (const int*)d_in[0];
    const float* feat    = (const float*)d_in[1];
    // d_in[2] = tlabel (arange(C)) -- algebraically eliminated
    const float* centers = (const float*)d_in[3];
    float* out = (float*)d_out;

    float* ws       = (float*)d_ws;
    float* inv_norm = ws;
    float* s        = ws + 2048;
    float* s_part   = ws + 4096;
    int*   cnt      = (int*)(ws + 36864);
    float* isl_part = ws + 38912;
    float* cl_part  = ws + 40960;

    init_cnt_k<<<(C_SZ + 255) / 256, 256, 0, stream>>>(cnt);
    rownorm_k<<<C_SZ, 256, 0, stream>>>(centers, inv_norm);
    colsum_k<<<dim3(D_SZ / 256, 16, 1), 256, 0, stream>>>(centers, inv_norm, s_part);
    sreduce_k<<<D_SZ / 256, 256, 0, stream>>>(s_part, s);
    hist_k<<<B_SZ / 256, 256, 0, stream>>>(label, cnt);
    classdot_k<<<C_SZ / 16, 32, 0, stream>>>(centers, inv_norm, s, cnt, isl_part);
    closs_k<<<B_SZ, 256, 0, stream>>>(feat, centers, label, cl_part);
    final_k<<<1, 1024, 0, stream>>>(cl_part, isl_part, out);
}